// MetaSignatureEncoder_20212116095301
// MI455X (gfx1250) — compile-verified
//
#include <hip/hip_runtime.h>
#include <hip/hip_bf16.h>

// ---------------------------------------------------------------------------
// MetaSignatureEncoder for MI455X (gfx1250, wave32).
// GEMMs run on V_WMMA_F32_16X16X4_F32 (exact fp32, matches reference dtype).
// Edge scatter uses native global_atomic_add_f32; feature maps fit in L2.
// ---------------------------------------------------------------------------

typedef __attribute__((ext_vector_type(2))) float v2f;
typedef __attribute__((ext_vector_type(8))) float v8f;

#define IN_CH 256
#define HID   256
#define OUT_CH 128
#define LN_EPS 1e-5f

// ---------------- degree / norm -------------------------------------------

__global__ void init_deg(float* deg, int n) {
  int i = blockIdx.x * blockDim.x + threadIdx.x;
  if (i < n) deg[i] = 1.0f;            // self loop contributes 1 to deg(src)
}

__global__ void deg_accum(const long long* __restrict__ ei, float* deg, int E) {
  int e = blockIdx.x * blockDim.x + threadIdx.x;
  if (e < E) unsafeAtomicAdd(&deg[(int)ei[e]], 1.0f);   // src row of edge_index
}

__global__ void deg_to_dinv(float* deg, int n) {
  int i = blockIdx.x * blockDim.x + threadIdx.x;
  if (i < n) deg[i] = rsqrtf(deg[i]);  // deg >= 1 always (self loops)
}

// ---------------- WMMA fp32 GEMM:  Y[MxKout] = X[MxK] @ W[KxKout] ----------
// One wave computes a 16x16 tile. 8 waves per block -> 128 output columns.
// A 16x4 layout (ISA 7.12.2): lanes 0-15 hold K={0,1}, lanes 16-31 K={2,3}.
// B 4x16 mirrored. C/D: VGPR r -> M = r + 8*(lane>=16), N = lane&15.

__global__ __launch_bounds__(256) void gemm_wmma_f32(
    const float* __restrict__ X, const float* __restrict__ W,
    float* __restrict__ Y, int M, int K, int Kout) {
  const int lane = threadIdx.x & 31;
  const int wave = threadIdx.x >> 5;
  const int m0 = blockIdx.x * 16;
  const int n0 = (blockIdx.y * 8 + wave) * 16;
  if (m0 >= M || n0 >= Kout) return;       // wave-uniform: EXEC stays all-ones

  const int mrow = m0 + (lane & 15);
  const int ncol = n0 + (lane & 15);
  const int koff = (lane >> 4) * 2;        // 0 for lanes 0-15, 2 for 16-31

  v8f acc = {};
  for (int k = 0; k < K; k += 4) {
    v2f a, b;
    const float* ap = X + (size_t)mrow * K + (k + koff);
    a[0] = ap[0];
    a[1] = ap[1];
    const float* bp = W + (size_t)(k + koff) * Kout + ncol;
    b[0] = bp[0];
    b[1] = bp[Kout];
    acc = __builtin_amdgcn_wmma_f32_16x16x4_f32(
        /*neg_a=*/false, a, /*neg_b=*/false, b,
        /*c_mod=*/(short)0, acc, /*reuse_a=*/false, /*reuse_b=*/false);
  }

  const int rbase = (lane >> 4) * 8;
  const int col = n0 + (lane & 15);
#pragma unroll
  for (int r = 0; r < 8; ++r)
    Y[(size_t)(m0 + rbase + r) * Kout + col] = acc[r];
}

// ---------------- propagation: out[dst] += dinv[s]*dinv[d] * h[src] --------

template <int C>
__global__ void self_prop(const float* __restrict__ dinv,
                          const float* __restrict__ h,
                          float* __restrict__ out, int n) {
  size_t idx = (size_t)blockIdx.x * blockDim.x + threadIdx.x;
  if (idx >= (size_t)n * C) return;
  int i = (int)(idx / C);
  float di = dinv[i];
  out[idx] = di * di * h[idx];             // initializes out + self-loop term
}

template <int C>
__global__ void edge_prop(const long long* __restrict__ ei,
                          const float* __restrict__ dinv,
                          const float* __restrict__ h,
                          float* __restrict__ out, int E) {
  int e = blockIdx.x * (blockDim.x >> 5) + (threadIdx.x >> 5);
  int lane = threadIdx.x & 31;
  if (e >= E) return;                      // wave-uniform exit
  int s = (int)ei[e];
  int d = (int)ei[(size_t)E + e];
  float w = dinv[s] * dinv[d];
  const float* hs = h + (size_t)s * C;
  float* od = out + (size_t)d * C;
#pragma unroll
  for (int j = 0; j < C / 32; ++j) {
    int f = lane + j * 32;                 // coalesced across the wave
    unsafeAtomicAdd(&od[f], w * hs[f]);
  }
}

// ---------------- graph signature ------------------------------------------

__global__ void zero_vec(float* v, int n) {
  int i = blockIdx.x * blockDim.x + threadIdx.x;
  if (i < n) v[i] = 0.0f;
}

__global__ void sig_reduce(const float* __restrict__ agg,
                           const float* __restrict__ bias,
                           float* __restrict__ s, int n) {
  int f = threadIdx.x;                     // 256 features
  float bf = bias[f];
  float acc = 0.0f;
  for (int i = blockIdx.x; i < n; i += gridDim.x)
    acc += fmaxf(agg[(size_t)i * HID + f] + bf, 0.0f);
  unsafeAtomicAdd(&s[f], acc);
}

// gamma1(256) | beta1(256) | gamma2(128) | beta2(128) packed in gb[768]
__global__ void fc_gammabeta(const float* __restrict__ s,
                             const float* __restrict__ w1, const float* __restrict__ b1,
                             const float* __restrict__ w2, const float* __restrict__ b2,
                             const float* __restrict__ w3, const float* __restrict__ b3,
                             const float* __restrict__ w4, const float* __restrict__ b4,
                             float* __restrict__ gb) {
  __shared__ float sl[HID];
  sl[threadIdx.x] = s[threadIdx.x];
  __syncthreads();
  int t = threadIdx.x;
  const float* w; const float* b; int j; int off;
  if (blockIdx.x == 0)      { w = w1; b = b1; j = t;        off = 0;   }
  else if (blockIdx.x == 1) { w = w2; b = b2; j = t;        off = 256; }
  else {
    if (t < 128) { w = w3; b = b3; j = t;       off = 512; }
    else         { w = w4; b = b4; j = t - 128; off = 640 - 128; }
  }
  float acc = 0.0f;
  const float* wr = w + (size_t)j * HID;   // fc*_w is [rows, HID], we need s @ W^T
#pragma unroll 8
  for (int k = 0; k < HID; ++k) acc += sl[k] * wr[k];
  gb[off + t] = tanhf(acc + b[j]);
}

// ---------------- FiLM + bias (+ReLU) + LayerNorm --------------------------

template <int C, int RELU>
__global__ void film_ln(const float* __restrict__ agg,
                        const float* __restrict__ gamma,
                        const float* __restrict__ beta,
                        const float* __restrict__ bias,
                        float* __restrict__ out, int n) {
  __shared__ float red[256];
  int i = blockIdx.x;
  int f = threadIdx.x;
  float v = gamma[f] * agg[(size_t)i * C + f] + beta[f] + bias[f];
  if (RELU) v = fmaxf(v, 0.0f);

  const float invC = 1.0f / (float)C;
  red[f] = v;
  __syncthreads();
  for (int off = C >> 1; off > 0; off >>= 1) {
    if (f < off) red[f] += red[f + off];
    __syncthreads();
  }
  float mu = red[0] * invC;
  __syncthreads();
  float d = v - mu;
  red[f] = d * d;
  __syncthreads();
  for (int off = C >> 1; off > 0; off >>= 1) {
    if (f < off) red[f] += red[f + off];
    __syncthreads();
  }
  float var = red[0] * invC;
  out[(size_t)i * C + f] = d * rsqrtf(var + LN_EPS);
}

// ---------------------------------------------------------------------------

extern "C" void kernel_launch(void* const* d_in, const int* in_sizes, int n_in,
                              void* d_out, int out_size, void* d_ws, size_t ws_size,
                              hipStream_t stream) {
  const float*     x        = (const float*)d_in[0];
  const long long* ei       = (const long long*)d_in[1];
  const float*     conv1_w  = (const float*)d_in[2];
  const float*     conv1_b  = (const float*)d_in[3];
  const float*     conv2_w  = (const float*)d_in[4];
  const float*     conv2_b  = (const float*)d_in[5];
  const float*     sig_w    = (const float*)d_in[6];
  const float*     sig_b    = (const float*)d_in[7];
  const float*     fc1_w    = (const float*)d_in[8];
  const float*     fc1_b    = (const float*)d_in[9];
  const float*     fc2_w    = (const float*)d_in[10];
  const float*     fc2_b    = (const float*)d_in[11];
  const float*     fc3_w    = (const float*)d_in[12];
  const float*     fc3_b    = (const float*)d_in[13];
  const float*     fc4_w    = (const float*)d_in[14];
  const float*     fc4_b    = (const float*)d_in[15];
  float* out = (float*)d_out;

  const int N = in_sizes[0] / IN_CH;    // 50000
  const int E = in_sizes[1] / 2;        // 800000

  // workspace layout (floats): dinv[N] | buf1[N*HID] | buf2[N*HID] | s[256] | gb[768]
  float* ws   = (float*)d_ws;
  float* dinv = ws;
  float* buf1 = dinv + N;
  float* buf2 = buf1 + (size_t)N * HID;
  float* sbuf = buf2 + (size_t)N * HID;
  float* gb   = sbuf + HID;

  const int mt = (N + 15) / 16;         // 3125 M-tiles

  // 1) symmetric GCN normalization
  init_deg<<<(N + 255) / 256, 256, 0, stream>>>(dinv, N);
  deg_accum<<<(E + 255) / 256, 256, 0, stream>>>(ei, dinv, E);
  deg_to_dinv<<<(N + 255) / 256, 256, 0, stream>>>(dinv, N);

  // 2) signature conv: h = x @ sig_w ; aggregate ; s = sum relu(agg + b)
  gemm_wmma_f32<<<dim3(mt, HID / 128), 256, 0, stream>>>(x, sig_w, buf1, N, IN_CH, HID);
  self_prop<HID><<<(int)(((size_t)N * HID + 255) / 256), 256, 0, stream>>>(dinv, buf1, buf2, N);
  edge_prop<HID><<<(E + 7) / 8, 256, 0, stream>>>(ei, dinv, buf1, buf2, E);
  zero_vec<<<1, 256, 0, stream>>>(sbuf, HID);
  sig_reduce<<<256, 256, 0, stream>>>(buf2, sig_b, sbuf, N);

  // 3) FiLM parameters
  fc_gammabeta<<<3, 256, 0, stream>>>(sbuf, fc1_w, fc1_b, fc2_w, fc2_b,
                                      fc3_w, fc3_b, fc4_w, fc4_b, gb);

  // 4) layer 1: h = LN(relu(gamma1 * agg(x @ conv1_w) + beta1 + conv1_b))
  gemm_wmma_f32<<<dim3(mt, HID / 128), 256, 0, stream>>>(x, conv1_w, buf1, N, IN_CH, HID);
  self_prop<HID><<<(int)(((size_t)N * HID + 255) / 256), 256, 0, stream>>>(dinv, buf1, buf2, N);
  edge_prop<HID><<<(E + 7) / 8, 256, 0, stream>>>(ei, dinv, buf1, buf2, E);
  film_ln<HID, 1><<<N, HID, 0, stream>>>(buf2, gb + 0, gb + 256, conv1_b, buf1, N);

  // 5) layer 2: out = LN(gamma2 * agg(h @ conv2_w) + beta2 + conv2_b)
  gemm_wmma_f32<<<dim3(mt, 1), 256, 0, stream>>>(buf1, conv2_w, buf2, N, HID, OUT_CH);
  self_prop<OUT_CH><<<(int)(((size_t)N * OUT_CH + 255) / 256), 256, 0, stream>>>(dinv, buf2, buf1, N);
  edge_prop<OUT_CH><<<(E + 7) / 8, 256, 0, stream>>>(ei, dinv, buf2, buf1, E);
  film_ln<OUT_CH, 0><<<N, OUT_CH, 0, stream>>>(buf1, gb + 512, gb + 640, conv2_b, out, N);
}